// GaussianLikelihoodLoss_59227599012532
// MI455X (gfx1250) — compile-verified
//
#include <hip/hip_runtime.h>

// Gaussian likelihood loss: mean over (b,t) of diff^T Sigma^{-1} diff + logdet(Sigma)
// B=1024, N=16, T=256. cov layout [B,N,N,T] => consecutive t contiguous.
// One thread per (b,t); per-lane in-register Cholesky of the 16x16 SPD matrix
// (lower triangle only -- cov is symmetric), fused forward solve, log-det from
// pivots. Memory bound: ~168MB read-once => ~7.2us floor at 23.3 TB/s.
// gfx1250 paths: global_load_async_to_lds_b128 (+s_wait_asynccnt) staging,
// NT-hint streaming loads, and V_WMMA_F32_16X16X4_F32 as the cross-lane
// reduction engine (wave32).

#define B_DIM 1024
#define N_DIM 16
#define T_DIM 256
#define NTRI  136  // 16*17/2

static __host__ __device__ constexpr int tri(int i, int j) { return i * (i + 1) / 2 + j; }

typedef __attribute__((address_space(3))) float lds_float;
typedef __attribute__((ext_vector_type(2))) float v2f;
typedef __attribute__((ext_vector_type(8))) float v8f;

// Wave-level sum of one float per lane via WMMA f32 16x16x4.
// A layout (32-bit A 16x4): lanes 0-15 hold M=0..15 {K=0 in vgpr0, K=1 in vgpr1},
// lanes 16-31 hold M=0..15 {K=2, K=3}. With A = {v, 0} and B = ones(4x16):
//   D[m][n] = v[m] + v[m+16]  (same for every n).
// C/D layout: lane n<16 vgpr r = D[r][n]; lane n>=16 vgpr r = D[r+8][n-16].
// So sum(c[0..7]) = sum over half the lanes; xor-16 shuffle adds the other half.
__device__ __forceinline__ float wave_sum_wmma(float v) {
  v2f a = {v, 0.0f};
  v2f ones = {1.0f, 1.0f};
  v8f c = {};
  c = __builtin_amdgcn_wmma_f32_16x16x4_f32(
      /*neg_a=*/false, a, /*neg_b=*/false, ones,
      /*c_mod=*/(short)0, c, /*reuse_a=*/false, /*reuse_b=*/false);
  float s = ((c[0] + c[1]) + (c[2] + c[3])) + ((c[4] + c[5]) + (c[6] + c[7]));
  s += __shfl_xor(s, 16, 32);  // combine the two half-wave partial sums
  return s;                    // full 32-lane sum, uniform across the wave
}

__global__ void __launch_bounds__(T_DIM) gll_main_kernel(
    const float* __restrict__ pred,
    const float* __restrict__ targ,
    const float* __restrict__ cov,
    float* __restrict__ partial) {
  __shared__ float s_pred[N_DIM * T_DIM];   // 16 KB
  __shared__ float s_targ[N_DIM * T_DIM];   // 16 KB
  __shared__ float s_wave[T_DIM / 32];      // 8 wave partials

  const int b = blockIdx.x;   // batch
  const int t = threadIdx.x;  // time (contiguous across lanes => coalesced)

  // ---- Stage pred/targ slices for this batch into LDS via CDNA5 async copies.
  {
    const float* gp = pred + (size_t)b * (N_DIM * T_DIM);
    const float* gt = targ + (size_t)b * (N_DIM * T_DIM);
#pragma unroll
    for (int r = 0; r < 4; ++r) {
      const int idx4 = (t + r * T_DIM) * 4;  // float index of this lane's 16B chunk
      unsigned lp = (unsigned)(__UINTPTR_TYPE__)(lds_float*)&s_pred[idx4];
      unsigned lt = (unsigned)(__UINTPTR_TYPE__)(lds_float*)&s_targ[idx4];
      const float* ap = gp + idx4;
      const float* at = gt + idx4;
      asm volatile("global_load_async_to_lds_b128 %0, %1, off"
                   :: "v"(lp), "v"(ap) : "memory");
      asm volatile("global_load_async_to_lds_b128 %0, %1, off"
                   :: "v"(lt), "v"(at) : "memory");
    }
    asm volatile("s_wait_asynccnt 0" ::: "memory");
  }
  __syncthreads();

  // ---- diff vector for this (b,t): column reads from LDS (lane t => no conflicts)
  float d[N_DIM];
#pragma unroll
  for (int n = 0; n < N_DIM; ++n)
    d[n] = s_pred[n * T_DIM + t] - s_targ[n * T_DIM + t];

  // ---- Load lower triangle of Sigma. Element (i,j,t) at ((i*16+j)*256 + t).
  // Per-lane base + immediate offsets; wave-coalesced 128B per load; NT hint
  // (read-once stream, keep it out of the 192MB L2).
  const float* cb = cov + (size_t)b * (N_DIM * N_DIM * T_DIM) + t;
  float a[NTRI];
#pragma unroll
  for (int i = 0; i < N_DIM; ++i)
#pragma unroll
    for (int j = 0; j <= i; ++j)
      a[tri(i, j)] = __builtin_nontemporal_load(cb + (i * N_DIM + j) * T_DIM);

  // ---- Fused Cholesky + forward solve + log-det (fully unrolled, registers only).
  float log2det = 0.0f;
  float quad    = 0.0f;
#pragma unroll
  for (int k = 0; k < N_DIM; ++k) {
    const float dkk = a[tri(k, k)];            // pivot = l_kk^2, >= 1 since Sigma >= I
    log2det += __builtin_amdgcn_logf(dkk);     // v_log_f32 (log2)
    float r = __builtin_amdgcn_rsqf(dkk);      // v_rsq_f32
    r = r * (1.5f - 0.5f * dkk * r * r);       // one Newton step

#pragma unroll
    for (int i = k + 1; i < N_DIM; ++i)        // scale column k -> l_ik
      a[tri(i, k)] *= r;

    const float yk = d[k] * r;                 // forward solve step
    quad += yk * yk;

#pragma unroll
    for (int j = k + 1; j < N_DIM; ++j) {      // rank-1 trailing update
      const float ljk = a[tri(j, k)];
      d[j] -= yk * ljk;
#pragma unroll
      for (int i = j; i < N_DIM; ++i)
        a[tri(i, j)] -= a[tri(i, k)] * ljk;
    }
  }
  const float val = quad + 0.6931471805599453f * log2det;  // ln(det)

  // ---- Reduction: WMMA wave-sum, then fixed-order combine of 8 wave partials.
  const float wsum = wave_sum_wmma(val);
  const int wave = t >> 5;
  const int lane = t & 31;
  if (lane == 0) s_wave[wave] = wsum;
  __syncthreads();
  if (t == 0) {
    float tot = 0.0f;
#pragma unroll
    for (int w = 0; w < T_DIM / 32; ++w) tot += s_wave[w];  // fixed order
    partial[b] = tot;
  }
}

__global__ void __launch_bounds__(256) gll_reduce_kernel(
    const float* __restrict__ partial, float* __restrict__ out) {
  __shared__ float s_wave[8];
  const int t = threadIdx.x;
  // fixed-order serial adds per lane: deterministic across replays
  float v = ((partial[t] + partial[t + 256]) + (partial[t + 512] + partial[t + 768]));
  const float wsum = wave_sum_wmma(v);
  if ((t & 31) == 0) s_wave[t >> 5] = wsum;
  __syncthreads();
  if (t == 0) {
    float tot = 0.0f;
#pragma unroll
    for (int w = 0; w < 8; ++w) tot += s_wave[w];
    out[0] = tot * (1.0f / ((float)B_DIM * (float)T_DIM));
  }
}

extern "C" void kernel_launch(void* const* d_in, const int* in_sizes, int n_in,
                              void* d_out, int out_size, void* d_ws, size_t ws_size,
                              hipStream_t stream) {
  const float* pred = (const float*)d_in[0];  // [B,N,T]
  const float* targ = (const float*)d_in[1];  // [B,N,T]
  const float* cov  = (const float*)d_in[2];  // [B,N,N,T]
  float* out = (float*)d_out;                 // scalar f32
  float* partial = (float*)d_ws;              // B_DIM floats of scratch

  gll_main_kernel<<<B_DIM, T_DIM, 0, stream>>>(pred, targ, cov, partial);
  gll_reduce_kernel<<<1, 256, 0, stream>>>(partial, out);
}